// Net_24232205484470
// MI455X (gfx1250) — compile-verified
//
#include <hip/hip_runtime.h>
#include <hip/hip_bf16.h>

typedef __attribute__((ext_vector_type(2))) float v2f;
typedef __attribute__((ext_vector_type(8))) float v8f;

#define F_IN 1433
#define HDIM 128
#define CDIM 7
#define KC   64

// ---------------- utility ----------------
__global__ void zero_f32(float* p, long n) {
    long i = blockIdx.x * (long)blockDim.x + threadIdx.x;
    long stride = (long)gridDim.x * blockDim.x;
    for (; i < n; i += stride) p[i] = 0.f;
}

__global__ void degree_count(const int* __restrict__ src, const int* __restrict__ dst,
                             int* deg_src, int* deg_dst, int E) {
    int e = blockIdx.x * blockDim.x + threadIdx.x;
    if (e < E) {
        atomicAdd(&deg_src[src[e]], 1);
        atomicAdd(&deg_dst[dst[e]], 1);
    }
}

__global__ void degree_norm(const int* __restrict__ deg_src, const int* __restrict__ deg_dst,
                            float* norm_src, float* norm_dst, int N) {
    int i = blockIdx.x * blockDim.x + threadIdx.x;
    if (i < N) {
        norm_src[i] = rsqrtf(fmaxf((float)deg_src[i], 1.f));
        norm_dst[i] = rsqrtf(fmaxf((float)deg_dst[i], 1.f));
    }
}

// ---------------- GEMM1: H1[N,128] = (X * norm_src) @ W1, fp32 WMMA ----------------
// Block: 256 threads = 8 waves. Block computes 16 rows x 128 cols; wave w owns cols [16w,16w+16).
__global__ __launch_bounds__(256) void gemm1_wmma(const float* __restrict__ X,
                                                  const float* __restrict__ W1,
                                                  const float* __restrict__ norm_src,
                                                  float* __restrict__ H1, int N) {
    __shared__ float As[16][KC + 1];       // +1 dword pad: column reads hit distinct banks
    __shared__ float Bs[KC][HDIM + 4];

    const int tid  = threadIdx.x;
    const int wave = tid >> 5;
    const int lane = tid & 31;
    const int half = lane >> 4;
    const int l    = lane & 15;
    const int rowBase = blockIdx.x * 16;   // N % 16 == 0 for this problem
    const int colBase = wave * 16;

    v8f acc = {};

    for (int kc0 = 0; kc0 < F_IN; kc0 += KC) {
        // stage A tile (norm_src applied, zero-pad K tail)
        for (int i = tid; i < 16 * KC; i += 256) {
            int r = i / KC, c = i - r * KC;
            int gk = kc0 + c;
            int gr = rowBase + r;
            float v = 0.f;
            if (gk < F_IN) v = X[(long)gr * F_IN + gk] * norm_src[gr];
            As[r][c] = v;
        }
        // stage B tile (zero-pad K tail)
        for (int i = tid; i < KC * HDIM; i += 256) {
            int r = i >> 7, c = i & (HDIM - 1);
            int gk = kc0 + r;
            Bs[r][c] = (gk < F_IN) ? W1[(long)gk * HDIM + c] : 0.f;
        }
        __syncthreads();

#pragma unroll
        for (int kk = 0; kk < KC; kk += 4) {
            v2f a, b;
            a.x = As[l][kk + 2 * half];                 // A: M=l, K=2*half
            a.y = As[l][kk + 2 * half + 1];             //    M=l, K=2*half+1
            b.x = Bs[kk + 2 * half][colBase + l];       // B: K=2*half,   N=l
            b.y = Bs[kk + 2 * half + 1][colBase + l];   //    K=2*half+1, N=l
            acc = __builtin_amdgcn_wmma_f32_16x16x4_f32(
                false, a, false, b, (short)0, acc, false, false);
        }
        __syncthreads();
    }

    // C/D layout: acc[i] holds (row = i + 8*half, col = l)
#pragma unroll
    for (int i = 0; i < 8; i++) {
        int gr = rowBase + i + 8 * half;
        H1[(long)gr * HDIM + colBase + l] = acc[i];
    }
}

// ---------------- edge scatter, 128-float payload: one wave32 per edge ----------------
__global__ void scatter1(const int* __restrict__ src, const int* __restrict__ dst,
                         const float* __restrict__ H1, float* AGG1, int E) {
    long idx = blockIdx.x * (long)blockDim.x + threadIdx.x;
    long e   = idx >> 5;
    int lane = (int)(idx & 31);
    if (e < E) {
        int s = src[e], d = dst[e];
        const float4 v = *(const float4*)(H1 + (long)s * HDIM + lane * 4);
        float* o = AGG1 + (long)d * HDIM + lane * 4;
        atomicAdd(o + 0, v.x);
        atomicAdd(o + 1, v.y);
        atomicAdd(o + 2, v.z);
        atomicAdd(o + 3, v.w);
    }
}

__global__ void relu_norm_bias(const float* __restrict__ AGG1, const float* __restrict__ norm_dst,
                               const float* __restrict__ b1, float* X1, int N) {
    long i = blockIdx.x * (long)blockDim.x + threadIdx.x;
    if (i < (long)N * HDIM) {
        int r = (int)(i >> 7);
        int c = (int)(i & (HDIM - 1));
        float v = AGG1[i] * norm_dst[r] + b1[c];
        X1[i] = v > 0.f ? v : 0.f;
    }
}

// ---------------- GEMM2: H2[N,7] = (X1 * norm_src) @ W2 ----------------
__global__ void gemm2(const float* __restrict__ X1, const float* __restrict__ W2,
                      const float* __restrict__ norm_src, float* __restrict__ H2, int N) {
    int r = blockIdx.x * blockDim.x + threadIdx.x;
    if (r >= N) return;
    float ns = norm_src[r];
    float acc[CDIM] = {};
    const float* xr = X1 + (long)r * HDIM;
    for (int k = 0; k < HDIM; k++) {
        float xv = xr[k] * ns;
#pragma unroll
        for (int j = 0; j < CDIM; j++) acc[j] += xv * W2[k * CDIM + j];
    }
#pragma unroll
    for (int j = 0; j < CDIM; j++) H2[(long)r * CDIM + j] = acc[j];
}

__global__ void scatter2(const int* __restrict__ src, const int* __restrict__ dst,
                         const float* __restrict__ H2, float* AGG2, int E) {
    int e = blockIdx.x * blockDim.x + threadIdx.x;
    if (e < E) {
        int s = src[e], d = dst[e];
#pragma unroll
        for (int j = 0; j < CDIM; j++)
            atomicAdd(&AGG2[(long)d * CDIM + j], H2[(long)s * CDIM + j]);
    }
}

__global__ void final_out(const float* __restrict__ AGG2, const float* __restrict__ norm_dst,
                          const float* __restrict__ b2, float* out, int N) {
    long i = blockIdx.x * (long)blockDim.x + threadIdx.x;
    if (i < (long)N * CDIM) {
        int r = (int)(i / CDIM);
        int j = (int)(i - (long)r * CDIM);
        out[i] = AGG2[i] * norm_dst[r] + b2[j];
    }
}

// ---------------- launch ----------------
extern "C" void kernel_launch(void* const* d_in, const int* in_sizes, int n_in,
                              void* d_out, int out_size, void* d_ws, size_t ws_size,
                              hipStream_t stream) {
    const float* X   = (const float*)d_in[0];
    const int*   src = (const int*)d_in[1];
    const int*   dst = (const int*)d_in[2];
    const float* W1  = (const float*)d_in[3];
    const float* b1  = (const float*)d_in[4];
    const float* W2  = (const float*)d_in[5];
    const float* b2  = (const float*)d_in[6];
    float* out = (float*)d_out;

    const int N = in_sizes[0] / F_IN;
    const int E = in_sizes[1];

    // workspace layout (4-byte units)
    int*   deg_src  = (int*)d_ws;                       // N
    int*   deg_dst  = deg_src + N;                      // N
    float* norm_src = (float*)(deg_dst + N);            // N
    float* norm_dst = norm_src + N;                     // N
    float* H1       = norm_dst + N;                     // N*128 (reused as X1)
    float* AGG1     = H1 + (size_t)N * HDIM;            // N*128
    float* H2       = AGG1 + (size_t)N * HDIM;          // N*7
    float* AGG2     = H2 + (size_t)N * CDIM;            // N*7
    (void)n_in; (void)out_size; (void)ws_size;

    zero_f32<<<256, 256, 0, stream>>>((float*)deg_src, 2L * N);
    degree_count<<<(E + 255) / 256, 256, 0, stream>>>(src, dst, deg_src, deg_dst, E);
    degree_norm<<<(N + 255) / 256, 256, 0, stream>>>(deg_src, deg_dst, norm_src, norm_dst, N);

    gemm1_wmma<<<N / 16, 256, 0, stream>>>(X, W1, norm_src, H1, N);

    zero_f32<<<2048, 256, 0, stream>>>(AGG1, (long)N * HDIM);
    {
        long t = (long)E * 32;
        scatter1<<<(unsigned)((t + 255) / 256), 256, 0, stream>>>(src, dst, H1, AGG1, E);
    }
    {
        long t = (long)N * HDIM;
        relu_norm_bias<<<(unsigned)((t + 255) / 256), 256, 0, stream>>>(AGG1, norm_dst, b1, H1, N);
    }

    gemm2<<<(N + 255) / 256, 256, 0, stream>>>(H1, W2, norm_src, H2, N);

    zero_f32<<<256, 256, 0, stream>>>(AGG2, (long)N * CDIM);
    scatter2<<<(E + 255) / 256, 256, 0, stream>>>(src, dst, H2, AGG2, E);
    {
        long t = (long)N * CDIM;
        final_out<<<(unsigned)((t + 255) / 256), 256, 0, stream>>>(AGG2, norm_dst, b2, out, N);
    }
}